// AttentionBlock_17557826306423
// MI455X (gfx1250) — compile-verified
//
#include <hip/hip_runtime.h>
#include <hip/hip_bf16.h>

typedef __attribute__((ext_vector_type(2))) float v2f;
typedef __attribute__((ext_vector_type(8))) float v8f;

// D = A(16x4 f32) * B(4x16 f32) + C(16x16 f32), full fp32 matrix op on gfx1250
#define WMMA_F32(a, b, c) \
  __builtin_amdgcn_wmma_f32_16x16x4_f32(false, (a), false, (b), (short)0, (c), false, false)

constexpr int kN  = 50000;  // rows of x
constexpr int kIN = 256;    // sIN
constexpr int kNy = 4096;   // rows of y
constexpr int kYD = 7;      // yDIM
constexpr int kP  = 64;     // sPROJ
constexpr int kRowTiles = kN / 16;  // 3125 (exact)

__device__ inline v8f v8f_zero() {
  v8f z = {0.f, 0.f, 0.f, 0.f, 0.f, 0.f, 0.f, 0.f};
  return z;
}

// ---------------------------------------------------------------------------
// Tiny y-side projections (4096x7 @ 7x64): one thread per output element.
// Stores YK transposed + K-pair interleaved  (pairs (k,k+1) adjacent per col n)
// and YV row-pair interleaved (pairs (n,n+1) adjacent per col c) so that WMMA
// B-fragments are single aligned 8-byte loads.
// ---------------------------------------------------------------------------
__global__ void prep_y(const float* __restrict__ y, const float* __restrict__ Wk,
                       const float* __restrict__ bk, const float* __restrict__ Wv,
                       const float* __restrict__ bv, float* __restrict__ YKTi,
                       float* __restrict__ YVi) {
  int gid = blockIdx.x * blockDim.x + threadIdx.x;
  if (gid >= kNy * kP) return;
  int n = gid >> 6;
  int c = gid & 63;
  float yy[kYD];
#pragma unroll
  for (int d = 0; d < kYD; ++d) yy[d] = y[n * kYD + d];
  float vk = bk[c], vv = bv[c];
#pragma unroll
  for (int d = 0; d < kYD; ++d) {
    vk = fmaf(yy[d], Wk[d * kP + c], vk);
    vv = fmaf(yy[d], Wv[d * kP + c], vv);
  }
  YKTi[(((c >> 1) * kNy + n) << 1) | (c & 1)] = vk;   // element (k=c, col n)
  YVi[(((n >> 1) * kP + c) << 1) | (n & 1)] = vv;     // element (row n, col c)
}

// Pair-interleave Wq so B-fragments of the XQ GEMM are contiguous 8B loads.
__global__ void prep_wq(const float* __restrict__ Wq, float* __restrict__ Wqi) {
  int gid = blockIdx.x * blockDim.x + threadIdx.x;
  if (gid >= kIN * kP) return;
  int k = gid >> 6;
  int c = gid & 63;
  Wqi[(((k >> 1) * kP + c) << 1) | (k & 1)] = Wq[k * kP + c];
}

// ---------------------------------------------------------------------------
// XQ = x @ Wq + bq   (50000x256 @ 256x64), one wave per 16x64 output tile.
// ---------------------------------------------------------------------------
__global__ __launch_bounds__(256) void xq_proj(const float* __restrict__ x,
                                               const float* __restrict__ Wqi,
                                               const float* __restrict__ bq,
                                               float* __restrict__ XQ) {
  int wid = threadIdx.x >> 5;
  int gwave = blockIdx.x * (blockDim.x >> 5) + wid;
  if (gwave >= kRowTiles) return;
  int lane = threadIdx.x & 31;
  int l16 = lane & 15;
  int hl = lane >> 4;
  int row0 = gwave * 16;

  v8f acc[4];
#pragma unroll
  for (int ct = 0; ct < 4; ++ct) {
    float b = bq[ct * 16 + l16];  // same column -> same bias for all 8 row slots
#pragma unroll
    for (int i = 0; i < 8; ++i) acc[ct][i] = b;
  }

  const float* xrow = x + (size_t)(row0 + l16) * kIN;
#pragma unroll 4
  for (int t = 0; t < kIN / 4; ++t) {
    int kk = 4 * t + 2 * hl;
    v2f a = *(const v2f*)(xrow + kk);  // A frag: row l16, K pair (kk,kk+1)
#pragma unroll
    for (int ct = 0; ct < 4; ++ct) {
      v2f b = *(const v2f*)(Wqi + (((kk >> 1) * kP + ct * 16 + l16) << 1));
      acc[ct] = WMMA_F32(a, b, acc[ct]);
    }
  }
#pragma unroll
  for (int ct = 0; ct < 4; ++ct)
#pragma unroll
    for (int i = 0; i < 8; ++i)
      XQ[(size_t)(row0 + i + 8 * hl) * kP + ct * 16 + l16] = acc[ct][i];
}

// ---------------------------------------------------------------------------
// Fused attention: one wave streams a 16-row block over all 4096 columns.
//   S = XQ_tile @ YK^T  (16 wmma), p = exp(S), r = relu(S)
//   eacc += p @ YV, racc += r @ YV  (32 wmma), stats as per-lane partials.
// p/r tiles round-trip through LDS (pitch 18 -> conflict-free) to convert the
// C-register layout into WMMA A-layout. No max-shift needed (logits < ~16).
// ---------------------------------------------------------------------------
__global__ __launch_bounds__(256) void attn(const float* __restrict__ XQ,
                                            const float* __restrict__ YKTi,
                                            const float* __restrict__ YVi,
                                            float* __restrict__ out) {
  __shared__ __align__(16) float lds[8][2][16 * 18];
  int wid = threadIdx.x >> 5;
  int gwave = blockIdx.x * (blockDim.x >> 5) + wid;
  if (gwave >= kRowTiles) return;
  int lane = threadIdx.x & 31;
  int l16 = lane & 15;
  int hl = lane >> 4;
  int row0 = gwave * 16;
  float* pl = &lds[wid][0][0];
  float* rl = &lds[wid][1][0];

  // Preload all XQ A-fragments for this 16x64 block (reused for 256 tiles).
  v2f aq[16];
#pragma unroll
  for (int t = 0; t < 16; ++t)
    aq[t] = *(const v2f*)(XQ + (size_t)(row0 + l16) * kP + 4 * t + 2 * hl);

  v8f eacc[4], racc[4];
#pragma unroll
  for (int ct = 0; ct < 4; ++ct) {
    eacc[ct] = v8f_zero();
    racc[ct] = v8f_zero();
  }
  float psum[8] = {0.f, 0.f, 0.f, 0.f, 0.f, 0.f, 0.f, 0.f};
  float rsum[8] = {0.f, 0.f, 0.f, 0.f, 0.f, 0.f, 0.f, 0.f};

  for (int j = 0; j < kNy / 16; ++j) {
    int n0 = j * 16;
    // ---- S tile: 16x16 logits via 16 f32 WMMAs over K=64 ----
    v8f s = v8f_zero();
#pragma unroll
    for (int t = 0; t < 16; ++t) {
      int kk = 4 * t + 2 * hl;
      v2f b = *(const v2f*)(YKTi + (((kk >> 1) * kNy + n0 + l16) << 1));
      s = WMMA_F32(aq[t], b, s);
    }
    // ---- exp / relu + per-lane stats, stage to LDS in row-major [16][18] ----
#pragma unroll
    for (int i = 0; i < 8; ++i) {
      float sv = s[i];
      float pv = __expf(sv);
      float rv = fmaxf(sv, 0.f);
      psum[i] += pv;
      rsum[i] += rv;
      int m = i + 8 * hl;
      pl[m * 18 + l16] = pv;
      rl[m * 18 + l16] = rv;
    }
    // ---- reload as A-fragments (in-order LDS within the wave) ----
    v2f ap[4], ar[4];
#pragma unroll
    for (int t2 = 0; t2 < 4; ++t2) {
      int kk2 = 4 * t2 + 2 * hl;
      ap[t2] = *(const v2f*)(pl + l16 * 18 + kk2);
      ar[t2] = *(const v2f*)(rl + l16 * 18 + kk2);
    }
    // ---- accumulate p@YV and r@YV (B frags shared between both) ----
#pragma unroll
    for (int ct = 0; ct < 4; ++ct) {
#pragma unroll
      for (int t2 = 0; t2 < 4; ++t2) {
        int kv = n0 + 4 * t2 + 2 * hl;  // even
        v2f b = *(const v2f*)(YVi + (((kv >> 1) * kP + ct * 16 + l16) << 1));
        eacc[ct] = WMMA_F32(ap[t2], b, eacc[ct]);
        racc[ct] = WMMA_F32(ar[t2], b, racc[ct]);
      }
    }
  }

  // Row sums: reduce across the 16-lane half (xor 1,2,4,8 stays in-half).
#pragma unroll
  for (int i = 0; i < 8; ++i) {
#pragma unroll
    for (int off = 1; off < 16; off <<= 1) {
      psum[i] += __shfl_xor(psum[i], off, 32);
      rsum[i] += __shfl_xor(rsum[i], off, 32);
    }
  }

#pragma unroll
  for (int i = 0; i < 8; ++i) {
    float inv_den = 1.0f / (1.0f + 0.1f * rsum[i]);
    float inv_es = 1.0f / psum[i];
#pragma unroll
    for (int ct = 0; ct < 4; ++ct) {
      float val = (0.1f * racc[ct][i] + eacc[ct][i] * inv_es) * inv_den;
      out[(size_t)(row0 + i + 8 * hl) * kP + ct * 16 + l16] = val;
    }
  }
}

// ---------------------------------------------------------------------------
extern "C" void kernel_launch(void* const* d_in, const int* in_sizes, int n_in,
                              void* d_out, int out_size, void* d_ws, size_t ws_size,
                              hipStream_t stream) {
  (void)in_sizes; (void)n_in; (void)out_size; (void)ws_size;
  const float* x  = (const float*)d_in[0];
  const float* y  = (const float*)d_in[1];
  const float* Wq = (const float*)d_in[2];
  const float* bq = (const float*)d_in[3];
  const float* Wk = (const float*)d_in[4];
  const float* bk = (const float*)d_in[5];
  const float* Wv = (const float*)d_in[6];
  const float* bv = (const float*)d_in[7];
  float* out = (float*)d_out;

  float* ws = (float*)d_ws;
  float* XQ   = ws;                          // N*64      = 3,200,000 f
  float* YKTi = XQ + (size_t)kN * kP;        // 64*Ny     =   262,144 f
  float* YVi  = YKTi + (size_t)kP * kNy;     // Ny*64     =   262,144 f
  float* Wqi  = YVi + (size_t)kNy * kP;      // 256*64    =    16,384 f
  // total workspace ~15 MB, every element rewritten each call

  prep_y<<<(kNy * kP + 255) / 256, 256, 0, stream>>>(y, Wk, bk, Wv, bv, YKTi, YVi);
  prep_wq<<<(kIN * kP + 255) / 256, 256, 0, stream>>>(Wq, Wqi);

  int blocks = (kRowTiles + 7) / 8;  // 8 waves per 256-thread block
  xq_proj<<<blocks, 256, 0, stream>>>(x, Wqi, bq, XQ);
  attn<<<blocks, 256, 0, stream>>>(XQ, YKTi, YVi, out);
}